// ClusterMerging_79121887527224
// MI455X (gfx1250) — compile-verified
//
#include <hip/hip_runtime.h>
#include <hip/hip_bf16.h>
#include <math.h>

typedef __attribute__((ext_vector_type(2))) float v2f;
typedef __attribute__((ext_vector_type(8))) float v8f;

namespace {
constexpr int B   = 8;
constexpr int C   = 192;
constexpr int K   = 784;
constexpr int Mm  = 16;
constexpr int NNC = 8;
constexpr int H   = 112;
constexpr int Wd  = 112;
constexpr int N   = H * Wd;      // 12544
constexpr int W2  = 56;
constexpr int N2  = W2 * W2;     // 3136
constexpr int HID = 64;          // C/3
constexpr int NOUT = 2 * C;      // 384
constexpr unsigned NHALF = (unsigned)B * (unsigned)N2 * 128u / 2u; // 1605632
}

__device__ __forceinline__ unsigned rotl32(unsigned x, int d) {
  return (x << d) | (x >> (32 - d));
}

// JAX threefry2x32-20 with key = jax.random.key(42) -> [0, 42]
__device__ __forceinline__ void threefry2x32(unsigned x0, unsigned x1,
                                             unsigned& o0, unsigned& o1) {
  const unsigned k0 = 0u, k1 = 42u;
  const unsigned ks[3] = {k0, k1, k0 ^ k1 ^ 0x1BD11BDAu};
  const int R[2][4] = {{13, 15, 26, 6}, {17, 29, 16, 24}};
  x0 += ks[0]; x1 += ks[1];
  #pragma unroll
  for (int blk = 0; blk < 5; ++blk) {
    #pragma unroll
    for (int r = 0; r < 4; ++r) {
      x0 += x1; x1 = rotl32(x1, R[blk & 1][r]); x1 ^= x0;
    }
    x0 += ks[(blk + 1) % 3];
    x1 += ks[(blk + 2) % 3] + (unsigned)(blk + 1);
  }
  o0 = x0; o1 = x1;
}

__device__ __forceinline__ float geluf(float x) {
  return 0.5f * x * (1.0f + erff(x * 0.70710678118654752f));
}

// ---------------------------------------------------------------- kernels ---

__global__ void k_scatter(const float* __restrict__ pos, int* __restrict__ img) {
  int t = blockIdx.x * blockDim.x + threadIdx.x;
  if (t >= B * N) return;
  int b = t / N, n = t - b * N;
  int x = (int)pos[(size_t)t * 2 + 0];
  int y = (int)pos[(size_t)t * 2 + 1];
  img[(size_t)b * N + y * Wd + x] = n;
}

__global__ __launch_bounds__(256) void k_layernorm(
    const float* __restrict__ feat, const float* __restrict__ g,
    const float* __restrict__ bb, float* __restrict__ out) {
  int wave = threadIdx.x >> 5, lane = threadIdx.x & 31;
  int row = blockIdx.x * 8 + wave;  // B*N rows, grid covers exactly
  const float* x = feat + (size_t)row * C;
  float v[6];
  float s = 0.f;
  #pragma unroll
  for (int i = 0; i < 6; ++i) { v[i] = x[lane + 32 * i]; s += v[i]; }
  #pragma unroll
  for (int o = 16; o > 0; o >>= 1) s += __shfl_xor(s, o, 32);
  float mu = s * (1.0f / C);
  float q = 0.f;
  #pragma unroll
  for (int i = 0; i < 6; ++i) { float d = v[i] - mu; q += d * d; }
  #pragma unroll
  for (int o = 16; o > 0; o >>= 1) q += __shfl_xor(q, o, 32);
  float inv = rsqrtf(q * (1.0f / C) + 1e-5f);
  float* y = out + (size_t)row * C;
  #pragma unroll
  for (int i = 0; i < 6; ++i) {
    int c = lane + 32 * i;
    y[c] = (v[i] - mu) * inv * g[c] + bb[c];
  }
}

__global__ void k_prep(const int* __restrict__ img, const float* __restrict__ pos,
                       const float* __restrict__ mask, int* __restrict__ idx_ds,
                       float* __restrict__ out_pos, float* __restrict__ out_mask) {
  int t = blockIdx.x * blockDim.x + threadIdx.x;
  if (t >= B * N2) return;
  int b = t / N2, j = t - b * N2;
  int r = j / W2, c = j - r * W2;
  int id = img[(size_t)b * N + (2 * r) * Wd + 2 * c];
  idx_ds[t] = id;
  float px = pos[((size_t)b * N + id) * 2 + 0];
  float py = pos[((size_t)b * N + id) * 2 + 1];
  out_pos[(size_t)t * 2 + 0] = floorf(px * 0.5f);
  out_pos[(size_t)t * 2 + 1] = floorf(py * 0.5f);
  out_mask[t] = mask[(size_t)b * N + id];
}

__global__ void k_neighbors(const int* __restrict__ idx_ds,
                            const int* __restrict__ ma,
                            const int* __restrict__ member_idx,
                            const float* __restrict__ cmask,
                            int* __restrict__ mem_cat) {
  int t = blockIdx.x * blockDim.x + threadIdx.x;
  if (t >= B * N2) return;
  int b = t / N2;
  int self = idx_ds[t];
  const int* arow = ma + ((size_t)b * N + self) * NNC;
  float s0 = -3.4e38f, s1 = -3.4e38f, s2 = -3.4e38f;
  int i0 = self, i1 = self, i2 = self;
  unsigned ebase = (unsigned)t * 128u;
  for (int nc = 0; nc < NNC; ++nc) {
    int cl = arow[nc];
    size_t rb = ((size_t)b * K + cl) * Mm;
    for (int mm = 0; mm < Mm; ++mm) {
      int mid = member_idx[rb + mm];
      float cv = cmask[rb + mm];
      if (mid == self) cv = 0.f;
      float pr = fmaxf(cv, 1e-5f);
      unsigned e = ebase + (unsigned)(nc * Mm + mm);
      unsigned i = (e < NHALF) ? e : (e - NHALF);
      unsigned r0, r1;
      threefry2x32(i, i + NHALF, r0, r1);
      unsigned bits = (e < NHALF) ? r0 : r1;
      float u = __uint_as_float((bits >> 9) | 0x3f800000u) - 1.0f;
      float gk = -logf(-logf(u + 1e-20f) + 1e-20f);
      float sc = logf(pr) + gk;
      if (sc > s0)      { s2=s1; i2=i1; s1=s0; i1=i0; s0=sc; i0=mid; }
      else if (sc > s1) { s2=s1; i2=i1; s1=sc; i1=mid; }
      else if (sc > s2) { s2=sc; i2=mid; }
    }
  }
  int* o = mem_cat + (size_t)t * 4;
  o[0] = i0; o[1] = i1; o[2] = i2; o[3] = self;
}

// Fused: gather neighbor features, pos/feat MLP weights, agg construction,
// and agg @ lin_W via V_WMMA_F32_16X16X4_F32. One WG = 16 points of one batch.
__global__ __launch_bounds__(256) void k_fused(
    const float* __restrict__ feat_n, const float* __restrict__ pos,
    const int* __restrict__ idx_ds, const int* __restrict__ mem_cat,
    const float* __restrict__ w1_W, const float* __restrict__ w1_b,
    const float* __restrict__ ln_g, const float* __restrict__ ln_b,
    const float* __restrict__ f1_W, const float* __restrict__ f1_b,
    const float* __restrict__ f2_W, const float* __restrict__ f2_b,
    const float* __restrict__ lin_W, const float* __restrict__ lin_b,
    float* __restrict__ out_feat) {
  constexpr int FS = 196;  // padded stride (196 % 64 == 4) to dodge bank conflicts
  __shared__ float s_feat[64 * FS];   // 4 neighbors x 16 points, 192 ch
  __shared__ float s_w[64 * 33];      // per (point,neighbor) 32 weights
  __shared__ float s_A[16 * FS];      // staged A-tile for the current K-chunk
  __shared__ int   s_cat[64];
  __shared__ float s_posn[64 * 2];
  __shared__ float s_posd[16 * 2];

  const int tid = threadIdx.x;
  const int b = blockIdx.y;
  const int tile = blockIdx.x;   // 0..195
  const int p0 = tile * 16;

  if (tid < 64)
    s_cat[tid] = mem_cat[((size_t)b * N2 + p0 + (tid >> 2)) * 4 + (tid & 3)];
  if (tid < 16) {
    int id = idx_ds[(size_t)b * N2 + p0 + tid];
    s_posd[tid * 2 + 0] = pos[((size_t)b * N + id) * 2 + 0];
    s_posd[tid * 2 + 1] = pos[((size_t)b * N + id) * 2 + 1];
  }
  __syncthreads();
  if (tid < 64) {
    int id = s_cat[tid];
    s_posn[tid * 2 + 0] = pos[((size_t)b * N + id) * 2 + 0];
    s_posn[tid * 2 + 1] = pos[((size_t)b * N + id) * 2 + 1];
  }
  {  // cooperative gather of 64 feature rows (192 f32 each) into LDS
    int r = tid >> 2, q = tid & 3;
    const float* src = feat_n + ((size_t)b * N + s_cat[r]) * C + q * 48;
    float* dst = s_feat + r * FS + q * 48;
    #pragma unroll 12
    for (int i = 0; i < 48; ++i) dst[i] = src[i];
  }
  __syncthreads();

  {  // per-(point,neighbor) gating weights: 4 threads per pair
    const int pm = tid >> 2, sub = tid & 3;
    const int selfr = ((pm >> 2) << 2) + 3;  // m=3 is the point itself
    float hh[16];
    #pragma unroll
    for (int i = 0; i < 16; ++i) hh[i] = f1_b[sub * 16 + i];
    for (int c = 0; c < C; ++c) {
      float fr = s_feat[pm * FS + c] - s_feat[selfr * FS + c];
      const float* wr = f1_W + c * HID + sub * 16;
      #pragma unroll
      for (int i = 0; i < 16; ++i) hh[i] += fr * wr[i];
    }
    float a2 = 0.f;
    #pragma unroll
    for (int i = 0; i < 16; ++i) a2 += geluf(hh[i]) * f2_W[sub * 16 + i];
    a2 += __shfl_xor(a2, 1, 32);
    a2 += __shfl_xor(a2, 2, 32);
    if (sub == 0) {
      float fw = 1.0f / (1.0f + expf(-(a2 + f2_b[0])));
      int p = pm >> 2;
      float prx = s_posn[pm * 2 + 0] - s_posd[p * 2 + 0];
      float pry = s_posn[pm * 2 + 1] - s_posd[p * 2 + 1];
      float tv[32]; float mu = 0.f;
      #pragma unroll
      for (int j = 0; j < 32; ++j) {
        tv[j] = prx * w1_W[j] + pry * w1_W[32 + j] + w1_b[j];
        mu += tv[j];
      }
      mu *= (1.0f / 32.0f);
      float va = 0.f;
      #pragma unroll
      for (int j = 0; j < 32; ++j) { float d = tv[j] - mu; va += d * d; }
      float inv = rsqrtf(va * (1.0f / 32.0f) + 1e-5f);
      #pragma unroll
      for (int j = 0; j < 32; ++j) {
        float z = (tv[j] - mu) * inv * ln_g[j] + ln_b[j];
        s_w[pm * 33 + j] = geluf(z) * fw;
      }
    }
  }
  __syncthreads();

  // GEMM: out[16,384] = agg[16,6144] @ lin_W[6144,384], K split into 32
  // chunks of 192; A-tile regenerated per chunk from w and feat_g.
  const int lane = tid & 31, wave = tid >> 5;
  const int half = lane >> 4, lm = lane & 15;
  v8f acc0 = {}, acc1 = {}, acc2v = {};
  for (int kc = 0; kc < 32; ++kc) {
    for (int e = tid; e < 16 * C; e += 256) {
      int p = e / C, c = e - p * C;
      int base = p << 2;
      float a = s_w[(base + 0) * 33 + kc] * s_feat[(base + 0) * FS + c]
              + s_w[(base + 1) * 33 + kc] * s_feat[(base + 1) * FS + c]
              + s_w[(base + 2) * 33 + kc] * s_feat[(base + 2) * FS + c]
              + s_w[(base + 3) * 33 + kc] * s_feat[(base + 3) * FS + c];
      s_A[p * FS + c] = a;
    }
    __syncthreads();
    if (kc + 1 < 32)  // warm L2/near cache for next lin_W chunk
      __builtin_prefetch(lin_W + ((size_t)(kc + 1) * C + (tid & 63)) * NOUT, 0, 1);
    const int kbase = kc * C;
    for (int kk = 0; kk < C; kk += 4) {
      v2f a;
      a.x = s_A[lm * FS + kk + (half ? 2 : 0)];
      a.y = s_A[lm * FS + kk + (half ? 3 : 1)];
      const size_t krow = (size_t)(kbase + kk + (half ? 2 : 0)) * NOUT;
      {
        const float* wp = lin_W + krow + wave * 16 + lm;
        v2f bv; bv.x = wp[0]; bv.y = wp[NOUT];
        acc0 = __builtin_amdgcn_wmma_f32_16x16x4_f32(false, a, false, bv,
                                                     (short)0, acc0, false, false);
      }
      {
        const float* wp = lin_W + krow + (wave + 8) * 16 + lm;
        v2f bv; bv.x = wp[0]; bv.y = wp[NOUT];
        acc1 = __builtin_amdgcn_wmma_f32_16x16x4_f32(false, a, false, bv,
                                                     (short)0, acc1, false, false);
      }
      {
        const float* wp = lin_W + krow + (wave + 16) * 16 + lm;
        v2f bv; bv.x = wp[0]; bv.y = wp[NOUT];
        acc2v = __builtin_amdgcn_wmma_f32_16x16x4_f32(false, a, false, bv,
                                                      (short)0, acc2v, false, false);
      }
    }
    __syncthreads();
  }

  #pragma unroll
  for (int i = 0; i < 3; ++i) {
    v8f accv = (i == 0) ? acc0 : (i == 1) ? acc1 : acc2v;
    int nt = wave + 8 * i;
    float bias = lin_b[nt * 16 + lm];
    #pragma unroll
    for (int r = 0; r < 8; ++r) {
      int row = r + half * 8;  // D layout: VGPR r holds M=r (lanes 0-15), M=8+r
      out_feat[((size_t)b * N2 + p0 + row) * NOUT + nt * 16 + lm] = accv[r] + bias;
    }
  }
}

// ----------------------------------------------------------------- launch ---

extern "C" void kernel_launch(void* const* d_in, const int* in_sizes, int n_in,
                              void* d_out, int out_size, void* d_ws, size_t ws_size,
                              hipStream_t stream) {
  (void)in_sizes; (void)n_in; (void)out_size; (void)ws_size;
  const float* pos          = (const float*)d_in[0];
  const float* feat         = (const float*)d_in[1];
  const float* mask         = (const float*)d_in[2];
  const int*   mean_assign  = (const int*)d_in[4];
  const int*   member_idx   = (const int*)d_in[5];
  const float* cluster_mask = (const float*)d_in[9];
  const float* norm_g = (const float*)d_in[10];
  const float* norm_b = (const float*)d_in[11];
  const float* w1_W   = (const float*)d_in[12];
  const float* w1_b   = (const float*)d_in[13];
  const float* ln_g   = (const float*)d_in[14];
  const float* ln_b   = (const float*)d_in[15];
  const float* f1_W   = (const float*)d_in[16];
  const float* f1_b   = (const float*)d_in[17];
  const float* f2_W   = (const float*)d_in[18];
  const float* f2_b   = (const float*)d_in[19];
  const float* lin_W  = (const float*)d_in[20];
  const float* lin_b  = (const float*)d_in[21];

  float* feat_n = (float*)d_ws;                       // B*N*C f32
  int*   img    = (int*)(feat_n + (size_t)B * N * C); // B*N
  int*   idx_ds = img + (size_t)B * N;                // B*N2
  int*   mcat   = idx_ds + (size_t)B * N2;            // B*N2*4

  float* out_pos  = (float*)d_out;                    // B*N2*2
  float* out_feat = out_pos + (size_t)B * N2 * 2;     // B*N2*384
  float* out_mask = out_feat + (size_t)B * N2 * NOUT; // B*N2

  k_scatter<<<(B * N + 255) / 256, 256, 0, stream>>>(pos, img);
  k_layernorm<<<(B * N) / 8, 256, 0, stream>>>(feat, norm_g, norm_b, feat_n);
  k_prep<<<(B * N2 + 255) / 256, 256, 0, stream>>>(img, pos, mask, idx_ds,
                                                   out_pos, out_mask);
  k_neighbors<<<(B * N2 + 255) / 256, 256, 0, stream>>>(idx_ds, mean_assign,
                                                        member_idx, cluster_mask,
                                                        mcat);
  k_fused<<<dim3(N2 / 16, B), 256, 0, stream>>>(
      feat_n, pos, idx_ds, mcat, w1_W, w1_b, ln_g, ln_b,
      f1_W, f1_b, f2_W, f2_b, lin_W, lin_b, out_feat);
}